// SAGEConv_68143951118619
// MI455X (gfx1250) — compile-verified
//
#include <hip/hip_runtime.h>
#include <hip/hip_bf16.h>

typedef float v2f __attribute__((ext_vector_type(2)));
typedef float v8f __attribute__((ext_vector_type(8)));

#define C_CH 64   // channels (in == out == 64)

// ---------------------------------------------------------------------------
// K1: zero the aggregation buffer (we use d_out as the neigh_agg accumulator,
// then overwrite it in place in the GEMM kernel).
// ---------------------------------------------------------------------------
__global__ void sage_zero_kernel(float* __restrict__ p, int n4) {
    int i = blockIdx.x * blockDim.x + threadIdx.x;
    if (i < n4) ((float4*)p)[i] = make_float4(0.f, 0.f, 0.f, 0.f);
}

// ---------------------------------------------------------------------------
// K2: edge gather/scale/scatter-add.  16 lanes per edge, float4 per lane:
// the 64 channels of one edge are 256 contiguous bytes -> fully coalesced
// gather from x[col], and 64 global_atomic_add_f32 into agg[row].
// x (25.6 MB) and agg (25.6 MB) both live in the 192 MB L2, so the random
// gathers and the atomic RMWs are serviced by L2, not HBM.
// ---------------------------------------------------------------------------
__global__ __launch_bounds__(256)
void sage_edge_scatter_kernel(const float* __restrict__ x,
                              const long long* __restrict__ row,
                              const long long* __restrict__ col,
                              const float* __restrict__ ew,
                              float* __restrict__ agg, int E) {
    long long gid = (long long)blockIdx.x * blockDim.x + threadIdx.x;
    int e = (int)(gid >> 4);
    int q = (int)(gid & 15);
    if (e >= E) return;
    int r = (int)row[e];
    int c = (int)col[e];
    float w = ew[e];
    float4 v = ((const float4*)(x + (long long)c * C_CH))[q];
    float* dst = agg + (long long)r * C_CH + q * 4;
    unsafeAtomicAdd(dst + 0, v.x * w);   // global_atomic_add_f32
    unsafeAtomicAdd(dst + 1, v.y * w);
    unsafeAtomicAdd(dst + 2, v.z * w);
    unsafeAtomicAdd(dst + 3, v.w * w);
}

// ---------------------------------------------------------------------------
// K3: fused dual GEMM using V_WMMA_F32_16X16X4_F32 (full f32 precision — the
// GEMM is ~2% of total traffic, so there is no reason to drop to bf16/f16).
//   out = x @ Wself^T + agg @ Wneigh^T + bself + bneigh
// One wave per 16-row node tile; 8 waves / block.  Weights staged in LDS once
// per block.  In-place: agg == out; each wave loads its entire 16x64 agg tile
// into registers before storing its 16x64 out tile (disjoint tiles per wave).
//
// Fragment layouts (ISA 7.12.2, wave32):
//   A (16x4 f32): lane l -> M = l&15, VGPR0/1 = A[M][4*kb + 2*(l>>4) + {0,1}]
//   B (4x16 f32): lane l -> N = l&15, VGPR0/1 = B[4*kb + 2*(l>>4) + {0,1}][N]
//                 with B[k][n] = W[n][k]  (out = x @ W^T), so the fragment is
//                 a contiguous float2 from row-major W — same pattern as A.
//   C/D (16x16 f32): VGPR r, lane l -> D[r + 8*(l>>4)][l&15]
// ---------------------------------------------------------------------------
__global__ __launch_bounds__(256)
void sage_gemm_kernel(const float* __restrict__ x,
                      const float* __restrict__ Wself,
                      const float* __restrict__ Wneigh,
                      const float* __restrict__ bself,
                      const float* __restrict__ bneigh,
                      float* outagg,   // agg on input, out on output (in place)
                      int nTiles, int N) {
    __shared__ float ldsW[2 * C_CH * C_CH];   // 32 KB: Wself then Wneigh

    const int tid = threadIdx.x;
    // cooperative weight staging (float4, fully coalesced)
    {
        const float4* ws4 = (const float4*)Wself;
        const float4* wn4 = (const float4*)Wneigh;
        float4* d0 = (float4*)ldsW;
        float4* d1 = (float4*)(ldsW + C_CH * C_CH);
        #pragma unroll
        for (int i = 0; i < (C_CH * C_CH / 4) / 256; ++i) {
            d0[tid + i * 256] = ws4[tid + i * 256];
            d1[tid + i * 256] = wn4[tid + i * 256];
        }
    }
    __syncthreads();

    const int wave = tid >> 5;
    const int lane = tid & 31;
    const int tile = blockIdx.x * 8 + wave;
    if (tile >= nTiles) return;          // wave-uniform: EXEC stays all-ones

    const int row0 = tile * 16;
    const int m    = lane & 15;          // A row / B,C column within tile
    const int kh   = lane >> 4;          // which K-pair (0 -> k0,k1 ; 1 -> k2,k3)

    int rowm = row0 + m;
    if (rowm >= N) rowm = N - 1;         // clamp loads; stores predicated below

    // Load all A fragments (x and agg) into registers BEFORE any store.
    v2f ax[16], ag[16];
    const float* xrow = x      + (long long)rowm * C_CH + 2 * kh;
    const float* grow = outagg + (long long)rowm * C_CH + 2 * kh;
    #pragma unroll
    for (int kb = 0; kb < 16; ++kb) {
        ax[kb] = *(const v2f*)(xrow + 4 * kb);
        ag[kb] = *(const v2f*)(grow + 4 * kb);
    }

    #pragma unroll
    for (int nt = 0; nt < 4; ++nt) {
        const int n0 = nt * 16;
        float b = bself[n0 + m] + bneigh[n0 + m];   // bias depends on col only
        v8f acc;
        #pragma unroll
        for (int r = 0; r < 8; ++r) acc[r] = b;

        const float* wsf = ldsW + (n0 + m) * C_CH + 2 * kh;          // Wself frag base
        const float* wnf = wsf + C_CH * C_CH;                        // Wneigh frag base
        #pragma unroll
        for (int kb = 0; kb < 16; ++kb) {
            v2f bs = *(const v2f*)(wsf + 4 * kb);                    // ds_load_b64
            acc = __builtin_amdgcn_wmma_f32_16x16x4_f32(
                      false, ax[kb], false, bs, (short)0, acc, false, false);
            v2f bn = *(const v2f*)(wnf + 4 * kb);
            acc = __builtin_amdgcn_wmma_f32_16x16x4_f32(
                      false, ag[kb], false, bn, (short)0, acc, false, false);
        }

        // store D: VGPR r, lane l -> row = row0 + r + 8*kh, col = n0 + (l&15)
        #pragma unroll
        for (int r = 0; r < 8; ++r) {
            int orow = row0 + r + 8 * kh;
            if (orow < N) outagg[(long long)orow * C_CH + n0 + m] = acc[r];
        }
    }
}

extern "C" void kernel_launch(void* const* d_in, const int* in_sizes, int n_in,
                              void* d_out, int out_size, void* d_ws, size_t ws_size,
                              hipStream_t stream) {
    const float*     x      = (const float*)d_in[0];
    const long long* eidx   = (const long long*)d_in[1];   // [2, E] int64
    const float*     ew     = (const float*)d_in[2];
    const float*     Wself  = (const float*)d_in[3];
    const float*     bself  = (const float*)d_in[4];
    const float*     Wneigh = (const float*)d_in[5];
    const float*     bneigh = (const float*)d_in[6];
    // d_in[7] = num_nodes scalar; derive N from out_size instead.

    const int E = in_sizes[2];             // edge_weight length
    const int N = out_size / C_CH;         // 100000
    float* out = (float*)d_out;

    // K1: zero agg accumulator (d_out doubles as agg buffer)
    const int n4 = (N * C_CH) / 4;
    sage_zero_kernel<<<(n4 + 255) / 256, 256, 0, stream>>>(out, n4);

    // K2: scatter-add messages (16 lanes / edge)
    const long long tThreads = (long long)E * 16;
    const int sblocks = (int)((tThreads + 255) / 256);
    sage_edge_scatter_kernel<<<sblocks, 256, 0, stream>>>(
        x, eidx, eidx + E, ew, out, E);

    // K3: fused dual WMMA GEMM, in place over d_out
    const int nTiles = (N + 15) / 16;
    sage_gemm_kernel<<<(nTiles + 7) / 8, 256, 0, stream>>>(
        x, Wself, Wneigh, bself, bneigh, out, nTiles, N);
}